// STDPAttentionMultiModal_67808943669997
// MI455X (gfx1250) — compile-verified
//
#include <hip/hip_runtime.h>
#include <hip/hip_bf16.h>

typedef __attribute__((ext_vector_type(2))) float v2f;
typedef __attribute__((ext_vector_type(4))) float v4f;
typedef __attribute__((ext_vector_type(8))) float v8f;

#define E_DIM   1024
#define N_SEQ   32
#define NN_DIM  4096
#define AUDIO_N 1024

// ---------------------------------------------------------------------------
// C[M x Nc] = A[M x K] @ W^T  (+ bias[n]),  W row-major [Nc x K].
// One wave32 computes one 16x16 tile via v_wmma_f32_16x16x4_f32, K step = 4.
// Grid must exactly tile (M/16)*(Nc/16); 8 waves per 256-thread block.
// ---------------------------------------------------------------------------
__global__ __launch_bounds__(256) void gemm_abt_wmma(
    const float* __restrict__ A,
    const float* __restrict__ W,      // [Nc x K] row-major, used as B = W^T
    const float* __restrict__ bias,   // [Nc] or nullptr
    float* __restrict__ C,
    int M, int Nc, int K)
{
    const int wave = threadIdx.x >> 5;           // wave32
    const int lane = threadIdx.x & 31;
    const int tile = blockIdx.x * 8 + wave;
    const int ntn  = Nc >> 4;
    const int tm   = tile / ntn;
    const int tn   = tile % ntn;

    const int half = lane >> 4;                  // 0: lanes 0-15, 1: lanes 16-31
    const int l15  = lane & 15;
    const int mg   = tm * 16 + l15;              // A row for this lane
    const int ng   = tn * 16 + l15;              // B/C column for this lane

    const float* arow = A + (size_t)mg * K + 2 * half;
    const float* brow = W + (size_t)ng * K + 2 * half;

    v8f acc = {};
#pragma unroll 4
    for (int k0 = 0; k0 < K; k0 += 4) {
        v2f a = *(const v2f*)(arow + k0);        // A[m][k0+2h], A[m][k0+2h+1]
        v2f b = *(const v2f*)(brow + k0);        // W^T[k0+2h][n], W^T[k0+2h+1][n]
        acc = __builtin_amdgcn_wmma_f32_16x16x4_f32(
            false, a, false, b, (short)0, acc, false, false);
    }

    const float bv = bias ? bias[ng] : 0.0f;
#pragma unroll
    for (int v = 0; v < 8; ++v) {
        // VGPR v holds row M = v + 8*half of the 16x16 tile
        C[(size_t)(tm * 16 + v + 8 * half) * Nc + ng] = acc[v] + bv;
    }
}

// ---------------------------------------------------------------------------
// C[M x Nc] = A[M x K] @ B[:, :Nc],  B row-major [K x ldb] (ldb >= Nc).
// Same WMMA tiling; B fragment needs two strided b32 loads per step.
// ---------------------------------------------------------------------------
__global__ __launch_bounds__(256) void gemm_ab_wmma(
    const float* __restrict__ A,
    const float* __restrict__ B,
    float* __restrict__ C,
    int M, int Nc, int K, int ldb)
{
    const int wave = threadIdx.x >> 5;
    const int lane = threadIdx.x & 31;
    const int tile = blockIdx.x * 8 + wave;
    const int ntn  = Nc >> 4;
    const int tm   = tile / ntn;
    const int tn   = tile % ntn;

    const int half = lane >> 4;
    const int l15  = lane & 15;
    const int mg   = tm * 16 + l15;
    const int ng   = tn * 16 + l15;

    const float* arow = A + (size_t)mg * K + 2 * half;
    const float* bcol = B + (size_t)(2 * half) * ldb + ng;

    v8f acc = {};
#pragma unroll 4
    for (int k0 = 0; k0 < K; k0 += 4) {
        v2f a = *(const v2f*)(arow + k0);
        v2f b;
        b.x = bcol[(size_t)k0 * ldb];            // B[k0+2h][n]
        b.y = bcol[(size_t)(k0 + 1) * ldb];      // B[k0+2h+1][n]
        acc = __builtin_amdgcn_wmma_f32_16x16x4_f32(
            false, a, false, b, (short)0, acc, false, false);
    }

#pragma unroll
    for (int v = 0; v < 8; ++v) {
        C[(size_t)(tm * 16 + v + 8 * half) * Nc + ng] = acc[v];
    }
}

// ---------------------------------------------------------------------------
// 16-byte vectorized grid-stride copy (the bandwidth-dominant part: 32 MB of
// weight pass-through since the STDP deltas are analytically zero).
// global_prefetch_b8 hint on the read stream (gfx1250 path).
// ---------------------------------------------------------------------------
__global__ __launch_bounds__(256) void copy_b128(
    const v4f* __restrict__ src, v4f* __restrict__ dst, int n4)
{
    int i = blockIdx.x * blockDim.x + threadIdx.x;
    const int stride = gridDim.x * blockDim.x;
    for (; i < n4; i += stride) {
        __builtin_prefetch(src + i + stride, 0, 0);   // -> global_prefetch_b8
        dst[i] = src[i];
    }
}

extern "C" void kernel_launch(void* const* d_in, const int* in_sizes, int n_in,
                              void* d_out, int out_size, void* d_ws, size_t ws_size,
                              hipStream_t stream) {
    (void)in_sizes; (void)n_in; (void)out_size; (void)ws_size;

    const float* av_value    = (const float*)d_in[0];
    // d_in[1] (av_key), d_in[2] (av_query): dead — attention over a length-1
    // axis is identity (softmax of a singleton == 1.0).
    const float* in_proj_w   = (const float*)d_in[3];   // [3E x E]
    const float* in_proj_b   = (const float*)d_in[4];   // [3E]
    const float* out_proj_w  = (const float*)d_in[5];   // [E x E]
    const float* out_proj_b  = (const float*)d_in[6];   // [E]
    const float* connections = (const float*)d_in[7];   // [E x NN]

    float* out       = (float*)d_out;
    float* audio     = out;                                      // 32*1024
    float* new_conn  = audio    + (size_t)N_SEQ * AUDIO_N;       // E*NN
    float* new_inw   = new_conn + (size_t)E_DIM * NN_DIM;        // 3E*E
    float* new_outw  = new_inw  + (size_t)3 * E_DIM * E_DIM;     // E*E

    float* vp   = (float*)d_ws;                 // [32 x 1024]
    float* attn = vp + (size_t)N_SEQ * E_DIM;   // [32 x 1024]

    // tiles = (32/16)*(1024/16) = 128 -> 16 blocks x 8 waves, exact cover.
    const int gemm_blocks = (N_SEQ / 16) * (E_DIM / 16) / 8;

    // vp = av_value @ Wv^T + bv   (Wv = in_proj_w[2E:3E])
    gemm_abt_wmma<<<gemm_blocks, 256, 0, stream>>>(
        av_value, in_proj_w + (size_t)2 * E_DIM * E_DIM,
        in_proj_b + 2 * E_DIM, vp, N_SEQ, E_DIM, E_DIM);

    // attn_out = vp @ Wo^T + bo
    gemm_abt_wmma<<<gemm_blocks, 256, 0, stream>>>(
        vp, out_proj_w, out_proj_b, attn, N_SEQ, E_DIM, E_DIM);

    // audio = attn_out @ connections[:, :AUDIO]  (rest of neuron_out is unused)
    gemm_ab_wmma<<<gemm_blocks, 256, 0, stream>>>(
        attn, connections, audio, N_SEQ, AUDIO_N, E_DIM, NN_DIM);

    // STDP deltas vanish (tau_pos == tau_neg) -> outputs are straight copies.
    copy_b128<<<1024, 256, 0, stream>>>(
        (const v4f*)connections, (v4f*)new_conn, (int)((size_t)E_DIM * NN_DIM / 4));
    copy_b128<<<1024, 256, 0, stream>>>(
        (const v4f*)in_proj_w, (v4f*)new_inw, (int)((size_t)3 * E_DIM * E_DIM / 4));
    copy_b128<<<1024, 256, 0, stream>>>(
        (const v4f*)out_proj_w, (v4f*)new_outw, (int)((size_t)E_DIM * E_DIM / 4));
}